// GNN_Encoder_37288906064247
// MI455X (gfx1250) — compile-verified
//
#include <hip/hip_runtime.h>

typedef __attribute__((ext_vector_type(2))) float v2f;
typedef __attribute__((ext_vector_type(8))) float v8f;

#define N_NODES 300000
#define DIM     128
#define NE1     300000
#define NT1     12000

// ---- init: deg=1 (self loop), cnt=0, s=0 -----------------------------------
__global__ void init_kernel(float* __restrict__ deg, float* __restrict__ cnt,
                            float* __restrict__ s) {
  int i = blockIdx.x * blockDim.x + threadIdx.x;
  if (i < NT1) { deg[i] = 1.0f; cnt[i] = 0.0f; }
  if (i < NT1 * DIM) s[i] = 0.0f;
}

// ---- deg[c] += attr[e_id[e]] -----------------------------------------------
__global__ void deg_kernel(const int* __restrict__ col, const int* __restrict__ eid,
                           const float* __restrict__ attr, float* __restrict__ deg) {
  int e = blockIdx.x * blockDim.x + threadIdx.x;
  if (e < NE1) atomicAdd(&deg[col[e]], attr[eid[e]]);
}

// ---- XW[m][j] = sum_k X[m][k]*W[j][k]  (X @ W^T), fp32 WMMA ----------------
// one block = 8 waves; wave w owns output tile (rows m0..m0+15, cols 16w..16w+15)
__global__ void gemm_xwT(const float* __restrict__ X, const float* __restrict__ W,
                         float* __restrict__ XW) {
  const int wave = threadIdx.x >> 5;
  const int lane = threadIdx.x & 31;
  const int half = lane >> 4;        // K sub-pair select
  const int l    = lane & 15;        // M (for A) / N (for B,C)
  const int m0   = blockIdx.x * 16;
  const int j0   = wave * 16;
  const float* arow = X + (size_t)(m0 + l) * DIM;
  const float* brow = W + (size_t)(j0 + l) * DIM;
  v8f acc = {};
#pragma unroll
  for (int k0 = 0; k0 < DIM; k0 += 4) {
    v2f a = *(const v2f*)(arow + k0 + 2 * half);   // A[M=l][K=k0+2h..+1]
    v2f b = *(const v2f*)(brow + k0 + 2 * half);   // B[K][N=l] = W[j0+l][k]
    acc = __builtin_amdgcn_wmma_f32_16x16x4_f32(false, a, false, b, (short)0, acc,
                                                false, false);
  }
#pragma unroll
  for (int i = 0; i < 8; ++i)
    XW[(size_t)(m0 + i + 8 * half) * DIM + j0 + l] = acc[i];
}

// ---- hT[i<T1] = b + xw[i]/deg[i]  (self-loop term) -------------------------
__global__ void hT_init(const float* __restrict__ xw, const float* __restrict__ deg,
                        const float* __restrict__ b, float* __restrict__ hT) {
  int i = blockIdx.x * blockDim.x + threadIdx.x;
  if (i < NT1 * DIM) hT[i] = b[i & (DIM - 1)] + xw[i] / deg[i >> 7];
}

// ---- GCN edge scatter: hT[c] += dinv_r*w*dinv_c * xw[r] --------------------
__global__ void gcn_edge(const int* __restrict__ row, const int* __restrict__ col,
                         const int* __restrict__ eid, const float* __restrict__ attr,
                         const float* __restrict__ deg, const float* __restrict__ xw,
                         float* __restrict__ hT) {
  int e = blockIdx.x, d = threadIdx.x;
  int c = col[e], r = row[e];
  float w  = attr[eid[e]];
  float dc = rsqrtf(deg[c]);
  float dr = (r < NT1) ? rsqrtf(deg[r]) : 1.0f;   // deg==1 for non-targets
  atomicAdd(&hT[(size_t)c * DIM + d], dr * w * dc * xw[(size_t)r * DIM + d]);
}

// ---- SAGE edge scatter: s[c] += h[r], cnt[c] += 1 --------------------------
__global__ void sage_edge(const int* __restrict__ row, const int* __restrict__ col,
                          const float* __restrict__ xw, const float* __restrict__ hT,
                          const float* __restrict__ b,
                          float* __restrict__ s, float* __restrict__ cnt) {
  int e = blockIdx.x, d = threadIdx.x;
  int c = col[e], r = row[e];
  float v = (r < NT1) ? hT[(size_t)r * DIM + d]
                      : (xw[(size_t)r * DIM + d] + b[d]);  // h[r>=T1] = xw+b
  atomicAdd(&s[(size_t)c * DIM + d], v);
  if (d == 0) atomicAdd(&cnt[c], 1.0f);
}

// ---- mean (in place) -------------------------------------------------------
__global__ void mean_kernel(float* __restrict__ s, const float* __restrict__ cnt) {
  int i = blockIdx.x * blockDim.x + threadIdx.x;
  if (i < NT1 * DIM) s[i] /= fmaxf(cnt[i >> 7], 1.0f);
}

// ---- Z = mean@Wl^T + hT@Wr^T + bl   (fp32 WMMA, bias in accumulator) -------
__global__ void gemm_out(const float* __restrict__ A1, const float* __restrict__ A2,
                         const float* __restrict__ B1, const float* __restrict__ B2,
                         const float* __restrict__ bias, float* __restrict__ Z) {
  const int wave = threadIdx.x >> 5;
  const int lane = threadIdx.x & 31;
  const int half = lane >> 4;
  const int l    = lane & 15;
  const int m0   = blockIdx.x * 16;
  const int j0   = wave * 16;
  const float* a1 = A1 + (size_t)(m0 + l) * DIM;
  const float* a2 = A2 + (size_t)(m0 + l) * DIM;
  const float* b1 = B1 + (size_t)(j0 + l) * DIM;
  const float* b2 = B2 + (size_t)(j0 + l) * DIM;
  float bl = bias[j0 + l];                       // bias depends on N only
  v8f acc = {bl, bl, bl, bl, bl, bl, bl, bl};
#pragma unroll
  for (int k0 = 0; k0 < DIM; k0 += 4) {
    v2f a = *(const v2f*)(a1 + k0 + 2 * half);
    v2f b = *(const v2f*)(b1 + k0 + 2 * half);
    acc = __builtin_amdgcn_wmma_f32_16x16x4_f32(false, a, false, b, (short)0, acc,
                                                false, false);
  }
#pragma unroll
  for (int k0 = 0; k0 < DIM; k0 += 4) {
    v2f a = *(const v2f*)(a2 + k0 + 2 * half);
    v2f b = *(const v2f*)(b2 + k0 + 2 * half);
    acc = __builtin_amdgcn_wmma_f32_16x16x4_f32(false, a, false, b, (short)0, acc,
                                                false, false);
  }
#pragma unroll
  for (int i = 0; i < 8; ++i)
    Z[(size_t)(m0 + i + 8 * half) * DIM + j0 + l] = acc[i];
}

// ---- row-wise L2 normalize, one wave32 per row -----------------------------
__global__ void norm_kernel(const float* __restrict__ Z, float* __restrict__ out) {
  int r    = blockIdx.x * 8 + (threadIdx.x >> 5);
  int lane = threadIdx.x & 31;
  float4 v = ((const float4*)(Z + (size_t)r * DIM))[lane];
  float ss = v.x * v.x + v.y * v.y + v.z * v.z + v.w * v.w;
#pragma unroll
  for (int off = 16; off > 0; off >>= 1) ss += __shfl_xor(ss, off, 32);
  float inv = 1.0f / fmaxf(sqrtf(ss), 1e-12f);
  float4 o = {v.x * inv, v.y * inv, v.z * inv, v.w * inv};
  ((float4*)(out + (size_t)r * DIM))[lane] = o;
}

extern "C" void kernel_launch(void* const* d_in, const int* in_sizes, int n_in,
                              void* d_out, int out_size, void* d_ws, size_t ws_size,
                              hipStream_t stream) {
  (void)in_sizes; (void)n_in; (void)out_size; (void)ws_size;
  const float* x     = (const float*)d_in[0];
  const float* attr  = (const float*)d_in[1];
  const float* gcn_W = (const float*)d_in[2];
  const float* gcn_b = (const float*)d_in[3];
  // layer-0 weights/edges (d_in[4..6,10..12,16]) are dead in the reference
  const float* s1_Wl = (const float*)d_in[7];
  const float* s1_bl = (const float*)d_in[8];
  const float* s1_Wr = (const float*)d_in[9];
  const int*   row1  = (const int*)d_in[13];
  const int*   col1  = (const int*)d_in[14];
  const int*   e_id1 = (const int*)d_in[15];

  float* ws   = (float*)d_ws;
  float* xw   = ws;                              // N*128
  float* hT   = xw  + (size_t)N_NODES * DIM;     // T1*128
  float* sbuf = hT  + (size_t)NT1 * DIM;         // T1*128
  float* z    = sbuf + (size_t)NT1 * DIM;        // T1*128
  float* deg  = z   + (size_t)NT1 * DIM;         // T1
  float* cnt  = deg + NT1;                       // T1
  float* out  = (float*)d_out;

  init_kernel<<<(NT1 * DIM + 255) / 256, 256, 0, stream>>>(deg, cnt, sbuf);
  deg_kernel <<<(NE1 + 255) / 256,       256, 0, stream>>>(col1, e_id1, attr, deg);
  gemm_xwT   <<<N_NODES / 16,            256, 0, stream>>>(x, gcn_W, xw);
  hT_init    <<<(NT1 * DIM + 255) / 256, 256, 0, stream>>>(xw, deg, gcn_b, hT);
  gcn_edge   <<<NE1,                     128, 0, stream>>>(row1, col1, e_id1, attr,
                                                           deg, xw, hT);
  sage_edge  <<<NE1,                     128, 0, stream>>>(row1, col1, xw, hT,
                                                           gcn_b, sbuf, cnt);
  mean_kernel<<<(NT1 * DIM + 255) / 256, 256, 0, stream>>>(sbuf, cnt);
  gemm_out   <<<NT1 / 16,                256, 0, stream>>>(sbuf, hT, s1_Wl, s1_Wr,
                                                           s1_bl, z);
  norm_kernel<<<NT1 / 8,                 256, 0, stream>>>(z, out);
}